// GAT_GCN_46969762349480
// MI455X (gfx1250) — compile-verified
//
#include <hip/hip_runtime.h>
#include <cmath>

// ---------------------------------------------------------------------------
// GAT_GCN on MI455X (gfx1250, wave32).
// Memory-bound workload: (1) fold the edge/attn projections so the 3.3M x 444
// 'et' tensor (5.9GB) is never materialized, (2) keep the 177.6MB node-feature
// matrices L2-resident (192MB L2) for gather/scatter, (3) exact-fp32 WMMA
// (V_WMMA_F32_16X16X4_F32), 32x64 macro-tile per wave: 2x M-tiles halve the
// dominant B-panel L2 traffic and give 8 independent WMMA chains. All load
// paths are branchless (clamped addresses + multiplicative masks) so the
// memory pipeline software-pipelines instead of saveexec/wait ladders.
// ---------------------------------------------------------------------------

static constexpr int FIN   = 37;
static constexpr int HEADS = 12;
static constexpr int HIDN  = FIN * HEADS;   // 444
static constexpr int EDIM  = 20;
static constexpr int GG    = 512;
static constexpr int FEATN = 1500;
static constexpr int OUTDN = 128;
static constexpr int FC1N  = 750;

typedef float v2f __attribute__((ext_vector_type(2)));
typedef float v8f __attribute__((ext_vector_type(8)));

// ---------------- small utility kernels ----------------

__global__ void fill_kernel(float* p, long n, float v) {
  long i = (long)blockIdx.x * blockDim.x + threadIdx.x;
  if (i < n) p[i] = v;
}

// dst[r,c] = bias[c]  (pre-seed accumulator rows with bias)
__global__ void init_rows_kernel(float* dst, const float* bias, long rows, int cols) {
  long i = (long)blockIdx.x * blockDim.x + threadIdx.x;
  if (i < rows * (long)cols) dst[i] = bias[i % cols];
}

__device__ inline void atomicMaxF(float* addr, float v) {
  // exact float max via ordered-int trick (init must be -inf = 0xFF800000)
  if (v >= 0.f) atomicMax((int*)addr, __float_as_int(v));
  else          atomicMin((unsigned int*)addr, __float_as_uint(v));
}

// ---------------- attention-projection folding ----------------
// Wsrc[f,h] = sum_c lin_w[f, h*37+c] * att_src[h,c]   (37x12)
// Wdst likewise; W2e[d,h] = sum_c lin_e_w[d, h*37+c] * att_edge[h,c] (20x12)
__global__ void fold_kernel(const float* lin_w, const float* att_src, const float* att_dst,
                            const float* lin_e_w, const float* att_edge,
                            float* Wsrc, float* Wdst, float* W2e) {
  int i = blockIdx.x * blockDim.x + threadIdx.x;
  if (i < FIN * HEADS) {
    int f = i / HEADS, h = i % HEADS;
    float s1 = 0.f, s2 = 0.f;
    for (int c = 0; c < FIN; ++c) {
      float w = lin_w[f * HIDN + h * FIN + c];
      s1 += w * att_src[h * FIN + c];
      s2 += w * att_dst[h * FIN + c];
    }
    Wsrc[i] = s1; Wdst[i] = s2;
  } else if (i < FIN * HEADS + EDIM * HEADS) {
    int j = i - FIN * HEADS;
    int d = j / HEADS, h = j % HEADS;
    float s = 0.f;
    for (int c = 0; c < FIN; ++c)
      s += lin_e_w[d * HIDN + h * FIN + c] * att_edge[h * FIN + c];
    W2e[j] = s;
  }
}

// a_src/a_dst = x @ Wsrc / x @ Wdst  (per node; folded weights cached in LDS)
__global__ void asrc_kernel(const float* x, const float* Wsrc, const float* Wdst,
                            float* a_src, float* a_dst, int N) {
  __shared__ float sW[2 * FIN * HEADS];
  for (int i = threadIdx.x; i < FIN * HEADS; i += blockDim.x) {
    sW[i] = Wsrc[i];
    sW[FIN * HEADS + i] = Wdst[i];
  }
  __syncthreads();
  int n = blockIdx.x * blockDim.x + threadIdx.x;
  if (n >= N) return;
  float xr[FIN];
  for (int f = 0; f < FIN; ++f) xr[f] = x[(long)n * FIN + f];
  for (int h = 0; h < HEADS; ++h) {
    float s1 = 0.f, s2 = 0.f;
    for (int f = 0; f < FIN; ++f) {
      s1 += xr[f] * sW[f * HEADS + h];
      s2 += xr[f] * sW[FIN * HEADS + f * HEADS + h];
    }
    a_src[(long)n * HEADS + h] = s1;
    a_dst[(long)n * HEADS + h] = s2;
  }
}

// ---------------- self-loop edge-attr mean ----------------
__global__ void deg_loopattr_kernel(const int* dst, const float* edge_attr,
                                    float* deg, float* loop_attr, int E) {
  int e = (blockIdx.x * blockDim.x + threadIdx.x) >> 5;
  int lane = threadIdx.x & 31;
  if (e >= E) return;
  int d = dst[e];
  if (lane == 0) atomicAdd(deg + d, 1.f);
  if (lane < EDIM) atomicAdd(loop_attr + (long)d * EDIM + lane, edge_attr[(long)e * EDIM + lane]);
}

__global__ void loopattr_norm_kernel(float* loop_attr, const float* deg, long n) {
  long i = (long)blockIdx.x * blockDim.x + threadIdx.x;
  if (i < n) loop_attr[i] /= fmaxf(deg[i / EDIM], 1.f);
}

// ---------------- alpha pass 1: leaky-relu logits + segment max ----------------
__global__ void alpha1_kernel(const int* src, const int* dst, const float* edge_attr,
                              const float* loop_attr, const float* W2e,
                              const float* a_src, const float* a_dst,
                              float* alpha, float* amax, int E, int N) {
  int e = (blockIdx.x * blockDim.x + threadIdx.x) >> 5;
  int lane = threadIdx.x & 31;
  if (e >= E + N) return;
  int s, d; const float* ea;
  if (e < E) { s = src[e]; d = dst[e]; ea = edge_attr + (long)e * EDIM; }
  else       { s = d = e - E;          ea = loop_attr + (long)(e - E) * EDIM; }
  if (lane < HEADS) {
    float ae = 0.f;
    for (int k = 0; k < EDIM; ++k) ae += ea[k] * W2e[k * HEADS + lane];
    float v = a_src[(long)s * HEADS + lane] + a_dst[(long)d * HEADS + lane] + ae;
    v = v > 0.f ? v : 0.2f * v;                       // leaky_relu(0.2)
    alpha[(long)e * HEADS + lane] = v;
    atomicMaxF(amax + (long)d * HEADS + lane, v);
  }
}

// ---------------- alpha pass 2: exp + segment sum ----------------
__global__ void alpha2_kernel(const int* dst, const float* amax,
                              float* alpha, float* asum, int E, int N) {
  long i = (long)blockIdx.x * blockDim.x + threadIdx.x;
  long total = (long)(E + N) * HEADS;
  if (i >= total) return;
  int e = (int)(i / HEADS), h = (int)(i % HEADS);
  int d = (e < E) ? dst[e] : e - E;
  float a = expf(alpha[i] - amax[(long)d * HEADS + h]);
  alpha[i] = a;
  atomicAdd(asum + (long)d * HEADS + h, a);
}

// ---------------- alpha pass 3: normalize + GAT aggregation ----------------
// h1[dst, h*37+c] += alpha[e,h] * xt[src, h*37+c]; xt (177.6MB) is L2-resident.
__global__ void alpha3_agg_kernel(const int* src, const int* dst, const float* asum,
                                  float* alpha, const float* xt, float* h1, int E, int N) {
  int e = (blockIdx.x * blockDim.x + threadIdx.x) >> 5;
  int lane = threadIdx.x & 31;
  if (e >= E + N) return;
  int s, d;
  if (e < E) { s = src[e]; d = dst[e]; } else { s = d = e - E; }
  float a = 0.f;
  if (lane < HEADS) {
    a = alpha[(long)e * HEADS + lane] / asum[(long)d * HEADS + lane];
    alpha[(long)e * HEADS + lane] = a;                // final alpha output
  }
  const float* xs = xt + (long)s * HIDN;
  float* hd = h1 + (long)d * HIDN;
  for (int j = lane; j < HIDN; j += 32) {
    float ah = __shfl(a, j / FIN, 32);
    atomicAdd(hd + j, ah * xs[j]);
  }
}

// ---------------- GCN norm + aggregation ----------------
__global__ void gcn_deg_kernel(const int* dst, const float* ew, float* deg, int E) {
  int e = blockIdx.x * blockDim.x + threadIdx.x;
  if (e < E) atomicAdd(deg + dst[e], ew[e]);
}

__global__ void dinv_kernel(float* deg, int N) {
  int n = blockIdx.x * blockDim.x + threadIdx.x;
  if (n >= N) return;
  float d = deg[n];
  deg[n] = d > 0.f ? 1.f / sqrtf(d) : 0.f;
}

__global__ void gcn_agg_kernel(const int* src, const int* dst, const float* ew,
                               const float* dinv, const float* xw, float* h2, int E, int N) {
  int e = (blockIdx.x * blockDim.x + threadIdx.x) >> 5;
  int lane = threadIdx.x & 31;
  if (e >= E + N) return;
  int s, d; float w;
  if (e < E) { s = src[e]; d = dst[e]; w = ew[e]; } else { s = d = e - E; w = 1.f; }
  float norm = dinv[s] * w * dinv[d];
  const float* xs = xw + (long)s * HIDN;
  float* hd = h2 + (long)d * HIDN;
  for (int j = lane; j < HIDN; j += 32) atomicAdd(hd + j, norm * xs[j]);
}

// ---------------- global max/mean pool (relu fused on load) ----------------
__global__ void pool_kernel(const int* batch, const float* h2, float* counts,
                            float* gmp, float* gap, int N) {
  int n = (blockIdx.x * blockDim.x + threadIdx.x) >> 5;
  int lane = threadIdx.x & 31;
  if (n >= N) return;
  int b = batch[n];
  if (lane == 0) atomicAdd(counts + b, 1.f);
  const float* hr = h2 + (long)n * HIDN;
  for (int j = lane; j < HIDN; j += 32) {
    float v = fmaxf(hr[j], 0.f);                      // relu; v >= 0
    atomicMax((int*)(gmp + (long)b * HIDN + j), __float_as_int(v)); // exact for v>=0, init 0
    atomicAdd(gap + (long)b * HIDN + j, v);
  }
}

__global__ void pool_fin_kernel(const float* gmp, const float* gap, const float* counts,
                                float* g) {
  int i = blockIdx.x * blockDim.x + threadIdx.x;
  if (i >= GG * HIDN) return;
  int b = i / HIDN, j = i % HIDN;
  g[(long)b * 2 * HIDN + j] = gmp[i];
  g[(long)b * 2 * HIDN + HIDN + j] = gap[i] / fmaxf(counts[b], 1.f);
}

// ---------------- fp32 WMMA GEMM: C = act(A @ B + bias) ----------------
// One wave32 per 32x64 macro-tile (2 M-tiles x 4 N-tiles): each B fragment is
// reused by 2 WMMAs (halves B-panel L2 traffic), each A fragment by 4, and the
// 8 accumulator chains are independent, covering XDL latency.
// K stepped by 4 via V_WMMA_F32_16X16X4_F32.
// Branchless everywhere: guarded path clamps indices and multiplies only the
// B fragment by a 0/1 K-mask (out-of-range M/N accumulator slots are simply
// never stored), so no load is ever under divergent EXEC and no
// saveexec/wait-per-load ladders appear. EXEC is all-1s at every WMMA.
// A frag (16x4): lanes 0-15 hold K=k0,k0+1; lanes 16-31 hold K=k0+2,k0+3.
// B frag (4x16): VGPR0 lanes 0-15 row k0 / lanes 16-31 row k0+2; VGPR1 +1.
// C/D (16x16 f32): VGPR r -> M=r (lanes 0-15) or M=8+r (lanes 16-31).
template <int RELUA>
__global__ void gemm_wmma_f32(const float* __restrict__ A, const float* __restrict__ B,
                              const float* __restrict__ bias, float* __restrict__ C,
                              int M, int N, int K, int reluOut) {
  int wave = blockIdx.x * (blockDim.x >> 5) + (threadIdx.x >> 5);
  int lane = threadIdx.x & 31;
  int tilesM2 = (M + 31) >> 5;
  int tilesN4 = (N + 63) >> 6;
  if (wave >= tilesM2 * tilesN4) return;              // wave-uniform exit
  int tm = wave / tilesN4, tn4 = wave - tm * tilesN4;
  int row0 = tm << 5, col0 = tn4 << 6;
  int half = lane >> 4, l16 = lane & 15;
  v8f acc[8];
#pragma unroll
  for (int t = 0; t < 8; ++t)
#pragma unroll
    for (int r = 0; r < 8; ++r) acc[t][r] = 0.f;

  const int K4 = K & ~3;
  const bool full = (row0 + 32 <= M) && (col0 + 64 <= N);

  if (full) {
    // ---- fast path: zero guards ----
    const float* Ap0 = A + (size_t)(row0 + l16) * K + half * 2;
    const float* Ap1 = Ap0 + (size_t)16 * K;
    const float* Bp  = B + (size_t)(half * 2) * N + col0 + l16;
    for (int k0 = 0; k0 < K4; k0 += 4) {
      v2f a0; a0[0] = Ap0[0]; a0[1] = Ap0[1];
      v2f a1; a1[0] = Ap1[0]; a1[1] = Ap1[1];
      if (RELUA) {
        a0[0] = a0[0] > 0.f ? a0[0] : 0.f; a0[1] = a0[1] > 0.f ? a0[1] : 0.f;
        a1[0] = a1[0] > 0.f ? a1[0] : 0.f; a1[1] = a1[1] > 0.f ? a1[1] : 0.f;
      }
#pragma unroll
      for (int t = 0; t < 4; ++t) {
        v2f bv; bv[0] = Bp[t * 16]; bv[1] = Bp[(size_t)N + t * 16];
        acc[t]     = __builtin_amdgcn_wmma_f32_16x16x4_f32(false, a0, false, bv,
                                                           (short)0, acc[t], false, false);
        acc[4 + t] = __builtin_amdgcn_wmma_f32_16x16x4_f32(false, a1, false, bv,
                                                           (short)0, acc[4 + t], false, false);
      }
      Ap0 += 4; Ap1 += 4; Bp += (size_t)4 * N;
    }
    if (K4 < K) {                                     // K tail: clamp + B-mask only
      int ka = K4 + half * 2;
      int k0c = ka < K ? ka : K - 1, k1c = ka + 1 < K ? ka + 1 : K - 1;
      float mk0 = ka < K ? 1.f : 0.f, mk1 = ka + 1 < K ? 1.f : 0.f;
      const float* Ar0 = A + (size_t)(row0 + l16) * K;
      const float* Ar1 = Ar0 + (size_t)16 * K;
      v2f a0; a0[0] = Ar0[k0c]; a0[1] = Ar0[k1c];
      v2f a1; a1[0] = Ar1[k0c]; a1[1] = Ar1[k1c];
      if (RELUA) {
        a0[0] = a0[0] > 0.f ? a0[0] : 0.f; a0[1] = a0[1] > 0.f ? a0[1] : 0.f;
        a1[0] = a1[0] > 0.f ? a1[0] : 0.f; a1[1] = a1[1] > 0.f ? a1[1] : 0.f;
      }
#pragma unroll
      for (int t = 0; t < 4; ++t) {
        int nB = col0 + t * 16 + l16;
        v2f bv; bv[0] = B[(size_t)k0c * N + nB] * mk0;
        bv[1] = B[(size_t)k1c * N + nB] * mk1;
        acc[t]     = __builtin_amdgcn_wmma_f32_16x16x4_f32(false, a0, false, bv,
                                                           (short)0, acc[t], false, false);
        acc[4 + t] = __builtin_amdgcn_wmma_f32_16x16x4_f32(false, a1, false, bv,
                                                           (short)0, acc[4 + t], false, false);
      }
    }
#pragma unroll
    for (int t = 0; t < 4; ++t) {
      int nB = col0 + t * 16 + l16;
      float bz = bias ? bias[nB] : 0.f;
#pragma unroll
      for (int r = 0; r < 8; ++r) {
        int m0 = row0 + half * 8 + r;
        float v0 = acc[t][r] + bz;
        float v1 = acc[4 + t][r] + bz;
        if (reluOut) { v0 = fmaxf(v0, 0.f); v1 = fmaxf(v1, 0.f); }
        C[(size_t)m0 * N + nB] = v0;
        C[(size_t)(m0 + 16) * N + nB] = v1;
      }
    }
  } else {
    // ---- guarded path: clamp indices; only B carries the K-mask; invalid
    // ---- M rows / N cols accumulate garbage into never-stored slots ----
    int r0c = row0 + l16;      r0c = r0c < M ? r0c : M - 1;
    int r1c = row0 + 16 + l16; r1c = r1c < M ? r1c : M - 1;
    const float* Ar0 = A + (size_t)r0c * K;
    const float* Ar1 = A + (size_t)r1c * K;
    for (int k0 = 0; k0 < K; k0 += 4) {
      int ka = k0 + half * 2;
      int k0c = ka < K ? ka : K - 1, k1c = ka + 1 < K ? ka + 1 : K - 1;
      float mk0 = ka < K ? 1.f : 0.f, mk1 = ka + 1 < K ? 1.f : 0.f;
      v2f a0; a0[0] = Ar0[k0c]; a0[1] = Ar0[k1c];
      v2f a1; a1[0] = Ar1[k0c]; a1[1] = Ar1[k1c];
      if (RELUA) {
        a0[0] = a0[0] > 0.f ? a0[0] : 0.f; a0[1] = a0[1] > 0.f ? a0[1] : 0.f;
        a1[0] = a1[0] > 0.f ? a1[0] : 0.f; a1[1] = a1[1] > 0.f ? a1[1] : 0.f;
      }
#pragma unroll
      for (int t = 0; t < 4; ++t) {
        int nB = col0 + t * 16 + l16; int nc = nB < N ? nB : N - 1;
        v2f bv; bv[0] = B[(size_t)k0c * N + nc] * mk0;
        bv[1] = B[(size_t)k1c * N + nc] * mk1;
        acc[t]     = __builtin_amdgcn_wmma_f32_16x16x4_f32(false, a0, false, bv,
                                                           (short)0, acc[t], false, false);
        acc[4 + t] = __builtin_amdgcn_wmma_f32_16x16x4_f32(false, a1, false, bv,
                                                           (short)0, acc[4 + t], false, false);
      }
    }
#pragma unroll
    for (int t = 0; t < 4; ++t) {
      int nB = col0 + t * 16 + l16;
      if (nB < N) {
        float bz = bias ? bias[nB] : 0.f;
#pragma unroll
        for (int r = 0; r < 8; ++r) {
          int m0 = row0 + half * 8 + r;
          if (m0 < M) {
            float v = acc[t][r] + bz;
            if (reluOut) v = fmaxf(v, 0.f);
            C[(size_t)m0 * N + nB] = v;
          }
          if (m0 + 16 < M) {
            float v = acc[4 + t][r] + bz;
            if (reluOut) v = fmaxf(v, 0.f);
            C[(size_t)(m0 + 16) * N + nB] = v;
          }
        }
      }
    }
  }
}

// ---------------- final 750->1 head with sigmoid ----------------
__global__ void head_out_kernel(const float* g3, const float* out_w, const float* out_b,
                                float* out, int G) {
  int g = (blockIdx.x * blockDim.x + threadIdx.x) >> 5;
  int lane = threadIdx.x & 31;
  if (g >= G) return;
  float s = 0.f;
  for (int j = lane; j < FC1N; j += 32) s += g3[(long)g * FC1N + j] * out_w[j];
  for (int off = 16; off > 0; off >>= 1) s += __shfl_xor(s, off, 32);
  if (lane == 0) out[g] = 1.f / (1.f + expf(-(s + out_b[0])));
}

// ---------------- host ----------------

static inline int cdiv(long a, long b) { return (int)((a + b - 1) / b); }

extern "C" void kernel_launch(void* const* d_in, const int* in_sizes, int n_in,
                              void* d_out, int out_size, void* d_ws, size_t ws_size,
                              hipStream_t stream) {
  const float* x          = (const float*)d_in[0];
  const int*   edge_index = (const int*)  d_in[1];
  const int*   batch      = (const int*)  d_in[2];
  const float* edge_attr  = (const float*)d_in[3];
  const float* edge_w     = (const float*)d_in[4];
  const float* gat_lin_w  = (const float*)d_in[5];
  const float* att_src    = (const float*)d_in[6];
  const float* att_dst    = (const float*)d_in[7];
  const float* lin_e_w    = (const float*)d_in[8];
  const float* att_edge   = (const float*)d_in[9];
  const float* gat_bias   = (const float*)d_in[10];
  const float* gcn_w      = (const float*)d_in[11];
  const float* gcn_b      = (const float*)d_in[12];
  const float* fc_g1_w    = (const float*)d_in[13];
  const float* fc_g1_b    = (const float*)d_in[14];
  const float* fc_g2_w    = (const float*)d_in[15];
  const float* fc_g2_b    = (const float*)d_in[16];
  const float* fc1_w      = (const float*)d_in[17];
  const float* fc1_b      = (const float*)d_in[18];
  const float* out_w      = (const float*)d_in[19];
  const float* out_b      = (const float*)d_in[20];

  const int E = in_sizes[1] / 2;
  const int N = in_sizes[2];
  const int* src = edge_index;
  const int* dst = edge_index + E;

  float* out_head  = (float*)d_out;        // [512]
  float* alpha_out = (float*)d_out + GG;   // [(E+N)*12], updated in place over 3 passes

  // workspace layout (xt aliased as xw after GAT; h1 aliased as h2 after GCN GEMM)
  float* w = (float*)d_ws;
  size_t off = 0;
  auto take = [&](size_t n) { float* p = w + off; off += n; return p; };
  float* xt    = take((size_t)N * HIDN);   // -> later reused as xw
  float* h1    = take((size_t)N * HIDN);   // -> later reused as h2
  float* a_src = take((size_t)N * HEADS);
  float* a_dst = take((size_t)N * HEADS);
  float* amax  = take((size_t)N * HEADS);
  float* asum  = take((size_t)N * HEADS);
  float* deg   = take((size_t)N);
  float* loopa = take((size_t)N * EDIM);
  float* dgcn  = take((size_t)N);          // deg then dinv in place
  float* Wsrc  = take(FIN * HEADS);
  float* Wdst  = take(FIN * HEADS);
  float* W2e   = take(EDIM * HEADS);
  float* cnts  = take(GG);
  float* gmp   = take((size_t)GG * HIDN);
  float* gap   = take((size_t)GG * HIDN);
  float* g     = take((size_t)GG * 2 * HIDN);
  float* g1    = take((size_t)GG * FEATN);
  float* g2    = take((size_t)GG * OUTDN);
  float* g3    = take((size_t)GG * FC1N);

  const int BT = 256;
  auto fill = [&](float* p, long n, float v) {
    fill_kernel<<<cdiv(n, BT), BT, 0, stream>>>(p, n, v);
  };
  auto gemm = [&](const float* A, const float* B, const float* bias, float* C,
                  int M, int Nn, int K, int reluA, int reluOut) {
    long waves = (long)((M + 31) / 32) * ((Nn + 63) / 64);
    int blocks = cdiv(waves, 8);
    if (reluA) gemm_wmma_f32<1><<<blocks, 256, 0, stream>>>(A, B, bias, C, M, Nn, K, reluOut);
    else       gemm_wmma_f32<0><<<blocks, 256, 0, stream>>>(A, B, bias, C, M, Nn, K, reluOut);
  };

  // init accumulators
  fill(deg, N, 0.f);
  fill(loopa, (long)N * EDIM, 0.f);
  fill(amax, (long)N * HEADS, -INFINITY);
  fill(asum, (long)N * HEADS, 0.f);
  fill(dgcn, N, 1.f);                      // self-loop weight 1
  fill(cnts, GG, 0.f);
  fill(gmp, (long)GG * HIDN, 0.f);         // relu'd features >= 0
  fill(gap, (long)GG * HIDN, 0.f);
  init_rows_kernel<<<cdiv((long)N * HIDN, BT), BT, 0, stream>>>(h1, gat_bias, N, HIDN);

  // folded attention projections + per-node logits
  fold_kernel<<<1, 768, 0, stream>>>(gat_lin_w, att_src, att_dst, lin_e_w, att_edge,
                                     Wsrc, Wdst, W2e);
  gemm(x, gat_lin_w, nullptr, xt, N, HIDN, FIN, 0, 0);          // xt = x @ lin_w (WMMA)
  asrc_kernel<<<cdiv(N, BT), BT, 0, stream>>>(x, Wsrc, Wdst, a_src, a_dst, N);

  // self-loop edge-attr mean
  deg_loopattr_kernel<<<cdiv((long)E * 32, BT), BT, 0, stream>>>(dst, edge_attr, deg, loopa, E);
  loopattr_norm_kernel<<<cdiv((long)N * EDIM, BT), BT, 0, stream>>>(loopa, deg, (long)N * EDIM);

  // scatter-softmax attention (3 passes over alpha, in the output buffer)
  alpha1_kernel<<<cdiv((long)(E + N) * 32, BT), BT, 0, stream>>>(
      src, dst, edge_attr, loopa, W2e, a_src, a_dst, alpha_out, amax, E, N);
  alpha2_kernel<<<cdiv((long)(E + N) * HEADS, BT), BT, 0, stream>>>(
      dst, amax, alpha_out, asum, E, N);
  alpha3_agg_kernel<<<cdiv((long)(E + N) * 32, BT), BT, 0, stream>>>(
      src, dst, asum, alpha_out, xt, h1, E, N);

  // GCN: norm, GEMM (relu fused on A load), aggregate
  gcn_deg_kernel<<<cdiv(E, BT), BT, 0, stream>>>(dst, edge_w, dgcn, E);
  dinv_kernel<<<cdiv(N, BT), BT, 0, stream>>>(dgcn, N);
  gemm(h1, gcn_w, nullptr, xt /*=xw*/, N, HIDN, HIDN, /*reluA*/1, 0);
  init_rows_kernel<<<cdiv((long)N * HIDN, BT), BT, 0, stream>>>(h1 /*=h2*/, gcn_b, N, HIDN);
  gcn_agg_kernel<<<cdiv((long)(E + N) * 32, BT), BT, 0, stream>>>(
      src, dst, edge_w, dgcn, xt /*xw*/, h1 /*h2*/, E, N);

  // pooling + MLP head (all GEMMs on WMMA path)
  pool_kernel<<<cdiv((long)N * 32, BT), BT, 0, stream>>>(batch, h1, cnts, gmp, gap, N);
  pool_fin_kernel<<<cdiv((long)GG * HIDN, BT), BT, 0, stream>>>(gmp, gap, cnts, g);
  gemm(g,  fc_g1_w, fc_g1_b, g1, GG, FEATN, 2 * HIDN, 0, 1);
  gemm(g1, fc_g2_w, fc_g2_b, g2, GG, OUTDN, FEATN,    0, 0);
  gemm(g2, fc1_w,   fc1_b,   g3, GG, FC1N,  OUTDN,    0, 1);
  head_out_kernel<<<cdiv((long)GG * 32, BT), BT, 0, stream>>>(g3, out_w, out_b, out_head, GG);
}